// HierarchicalSAE_65429531787656
// MI455X (gfx1250) — compile-verified
//
#include <hip/hip_runtime.h>

// Problem dims (fixed by the reference)
#define BB 4096   // batch
#define DD 2048   // model dim
#define PP 256    // parents
#define SS 64     // subspace
#define CC 32     // children per parent
#define KTOP 8    // active parents per row
#define BETA 0.1f

typedef float v2f __attribute__((ext_vector_type(2)));
typedef float v8f __attribute__((ext_vector_type(8)));

// CDNA5 fp32 WMMA: D(16x16,f32) = A(16x4,f32) x B(4x16,f32) + C
// Lane layout (ISA 7.12.2):
//   A: lane l holds A[l&15, k0 + 2*(l>>4) + {0,1}]
//   B: lane l holds B[k0 + 2*(l>>4) + {0,1}, l&15]
//   C/D: elem j of lane l is C[8*(l>>4)+j, l&15]
static __device__ __forceinline__ v8f wmma_f32x4(v2f a, v2f b, v8f c) {
  return __builtin_amdgcn_wmma_f32_16x16x4_f32(
      /*neg_a=*/false, a, /*neg_b=*/false, b,
      /*c_mod=*/(short)0, c, /*reuse_a=*/false, /*reuse_b=*/false);
}

// ---------------------------------------------------------------------------
// K0: zero per-parent counters
// ---------------------------------------------------------------------------
__global__ void k0_zero_counts(int* __restrict__ counts) {
  if (threadIdx.x < PP) counts[threadIdx.x] = 0;
}

// ---------------------------------------------------------------------------
// K1: parent_logits[B,P] = x @ W_re^T + b_re   (fp32 WMMA)
// grid = (BB/16, PP/64), block = 32 (one wave computes a 16x64 strip)
// ---------------------------------------------------------------------------
__global__ __launch_bounds__(32) void k1_parent_logits(
    const float* __restrict__ x, const float* __restrict__ W_re,
    const float* __restrict__ b_re, float* __restrict__ logits) {
  const int mt    = blockIdx.x;       // 16-row tile
  const int ng    = blockIdx.y;       // 64-col group
  const int lane  = threadIdx.x;
  const int mrow  = lane & 15;
  const int khalf = lane >> 4;

  const float* xrow = x + (size_t)(mt * 16 + mrow) * DD;
  const float* w0 = W_re + (size_t)(ng * 64 +  0 + mrow) * DD;
  const float* w1 = W_re + (size_t)(ng * 64 + 16 + mrow) * DD;
  const float* w2 = W_re + (size_t)(ng * 64 + 32 + mrow) * DD;
  const float* w3 = W_re + (size_t)(ng * 64 + 48 + mrow) * DD;

  v8f acc0 = {}; v8f acc1 = {}; v8f acc2 = {}; v8f acc3 = {};
  for (int k0 = 0; k0 < DD; k0 += 4) {
    const int kk = k0 + khalf * 2;
    v2f a  = *reinterpret_cast<const v2f*>(xrow + kk);
    v2f b0 = *reinterpret_cast<const v2f*>(w0 + kk);
    v2f b1 = *reinterpret_cast<const v2f*>(w1 + kk);
    v2f b2 = *reinterpret_cast<const v2f*>(w2 + kk);
    v2f b3 = *reinterpret_cast<const v2f*>(w3 + kk);
    acc0 = wmma_f32x4(a, b0, acc0);
    acc1 = wmma_f32x4(a, b1, acc1);
    acc2 = wmma_f32x4(a, b2, acc2);
    acc3 = wmma_f32x4(a, b3, acc3);
  }

  v8f accs[4] = {acc0, acc1, acc2, acc3};
  const int mbase = mt * 16 + khalf * 8;
#pragma unroll
  for (int nt = 0; nt < 4; ++nt) {
    const int n = ng * 64 + nt * 16 + mrow;
    const float bias = b_re[n];
#pragma unroll
    for (int j = 0; j < 8; ++j)
      logits[(size_t)(mbase + j) * PP + n] = accs[nt][j] + bias;
  }
}

// ---------------------------------------------------------------------------
// K2: per-row top-8 (jax tie-break: lowest index first) + per-parent lists
// ---------------------------------------------------------------------------
__global__ void k2_topk_and_lists(const float* __restrict__ logits,
                                  int* __restrict__ top_idx,
                                  int* __restrict__ counts,
                                  int* __restrict__ rowlist) {
  const int b = blockIdx.x * blockDim.x + threadIdx.x;
  if (b >= BB) return;
  const float* row = logits + (size_t)b * PP;
  int sel[KTOP];
#pragma unroll
  for (int t = 0; t < KTOP; ++t) {
    float best = -__builtin_inff();
    int bi = 0;
    for (int i = 0; i < PP; ++i) {
      bool taken = false;
#pragma unroll
      for (int u = 0; u < KTOP; ++u) taken = taken || (u < t && sel[u] == i);
      const float v = row[i];
      if (!taken && v > best) { best = v; bi = i; }
    }
    sel[t] = bi;
    top_idx[b * KTOP + t] = bi;
    const int pos = atomicAdd(&counts[bi], 1);
    rowlist[bi * BB + pos] = b * KTOP + t;   // pair id = b*8 + slot
  }
}

// ---------------------------------------------------------------------------
// K3: per (parent, 16-row tile):
//   (a) x_down = X_gathered @ W_down[p]^T            (WMMA, K=2048)
//   (b) child argmax over 32 children                (VALU + shfl merge)
//   (c) sub_recon rows = selected W_cd columns       (-> LDS A tile + subrec)
//   (d) if `partial`: child_recon tile = sub_recon @ W_up[p]^T (WMMA, N=2048)
//       scattered to pair-indexed slots (no fp atomics -> deterministic)
// grid = (PP, BB/16), block = 32. Blocks beyond the parent's count exit.
// ---------------------------------------------------------------------------
__global__ __launch_bounds__(32) void k3_child_encode(
    const float* __restrict__ x, const float* __restrict__ W_down,
    const float* __restrict__ W_ce, const float* __restrict__ b_ce,
    const float* __restrict__ W_cd, const float* __restrict__ W_up,
    const int* __restrict__ counts, const int* __restrict__ rowlist,
    float* __restrict__ subrec, float* __restrict__ partial) {
  const int p    = blockIdx.x;
  const int tile = blockIdx.y;
  const int n_p  = counts[p];
  if (tile * 16 >= n_p) return;          // uniform exit: EXEC stays all-ones

  const int lane  = threadIdx.x;
  const int mrow  = lane & 15;
  const int khalf = lane >> 4;

  const int pos  = tile * 16 + mrow;
  const int cpos = (pos < n_p) ? pos : (n_p - 1);   // clamp: dup row, no write
  const int pair = rowlist[p * BB + cpos];
  const float* xrow = x + (size_t)(pair >> 3) * DD;
  const float* wd   = W_down + (size_t)p * SS * DD;

  // (a) x_down tile [16 rows x 64 subspace] = X @ W_down[p]^T
  v8f acc0 = {}; v8f acc1 = {}; v8f acc2 = {}; v8f acc3 = {};
  for (int k0 = 0; k0 < DD; k0 += 4) {
    const int kk = k0 + khalf * 2;
    v2f a  = *reinterpret_cast<const v2f*>(xrow + kk);
    v2f b0 = *reinterpret_cast<const v2f*>(wd + (size_t)( 0 + mrow) * DD + kk);
    v2f b1 = *reinterpret_cast<const v2f*>(wd + (size_t)(16 + mrow) * DD + kk);
    v2f b2 = *reinterpret_cast<const v2f*>(wd + (size_t)(32 + mrow) * DD + kk);
    v2f b3 = *reinterpret_cast<const v2f*>(wd + (size_t)(48 + mrow) * DD + kk);
    acc0 = wmma_f32x4(a, b0, acc0);
    acc1 = wmma_f32x4(a, b1, acc1);
    acc2 = wmma_f32x4(a, b2, acc2);
    acc3 = wmma_f32x4(a, b3, acc3);
  }

  __shared__ float xds[16][SS + 1];      // +1 pad: dodge bank conflicts
  __shared__ float subA[16][SS];         // unpadded: v2f-aligned rows
  __shared__ int pairIds[16];
  {
    v8f accs[4] = {acc0, acc1, acc2, acc3};
#pragma unroll
    for (int nt = 0; nt < 4; ++nt)
#pragma unroll
      for (int j = 0; j < 8; ++j)
        xds[khalf * 8 + j][nt * 16 + mrow] = accs[nt][j];
  }
  __syncthreads();

  // (b) child logits + argmax. lane pair (l, l^16) splits the 32 children.
  const float* wce = W_ce + (size_t)p * CC * SS;
  float best = -__builtin_inff();
  int bestc = 0;
  const int c0 = khalf * 16;
  for (int c = c0; c < c0 + 16; ++c) {
    float sacc = b_ce[p * CC + c];
    const float* w = wce + (size_t)c * SS;
#pragma unroll 8
    for (int s = 0; s < SS; ++s) sacc += xds[mrow][s] * w[s];
    if (sacc > best) { best = sacc; bestc = c; }   // strict > : lowest idx wins
  }
  const float obest = __shfl_xor(best, 16, 32);
  const int   oc    = __shfl_xor(bestc, 16, 32);
  if (obest > best || (obest == best && oc < bestc)) { best = obest; bestc = oc; }

  // (c) sub_recon rows: v[s] = W_cd[p, s, c*]; invalid rows zeroed in LDS.
  for (int r = 0; r < 16; ++r) {
    const int cst = __shfl(bestc, r, 32);
    const int pr  = __shfl(pair, r, 32);
    const bool valid = (tile * 16 + r) < n_p;
    const float v0 = valid ? W_cd[((size_t)p * SS + lane)      * CC + cst] : 0.f;
    const float v1 = valid ? W_cd[((size_t)p * SS + lane + 32) * CC + cst] : 0.f;
    subA[r][lane]      = v0;
    subA[r][lane + 32] = v1;
    if (lane == 0) pairIds[r] = valid ? pr : -1;
    if (valid) {
      float* dst = subrec + (size_t)pr * SS;
      dst[lane]      = v0;
      dst[lane + 32] = v1;
    }
  }
  __syncthreads();

  // (d) WMMA up-projection: [16 x 64] @ W_up[p]^T -> [16 x 2048]
  if (partial != nullptr) {
    v2f areg[16];                        // whole A tile, reused for all N
#pragma unroll
    for (int t = 0; t < 16; ++t)
      areg[t] = *reinterpret_cast<const v2f*>(&subA[mrow][4 * t + khalf * 2]);
    int prs[8];
#pragma unroll
    for (int j = 0; j < 8; ++j) prs[j] = pairIds[khalf * 8 + j];

    const float* wu = W_up + (size_t)p * DD * SS;
    for (int nb = 0; nb < DD; nb += 64) {      // 4 N-tiles per iter (ILP)
      v8f d0 = {}; v8f d1 = {}; v8f d2 = {}; v8f d3 = {};
#pragma unroll
      for (int t = 0; t < 16; ++t) {
        const int k = 4 * t + khalf * 2;
        v2f w0 = *reinterpret_cast<const v2f*>(wu + (size_t)(nb +  0 + mrow) * SS + k);
        v2f w1 = *reinterpret_cast<const v2f*>(wu + (size_t)(nb + 16 + mrow) * SS + k);
        v2f w2 = *reinterpret_cast<const v2f*>(wu + (size_t)(nb + 32 + mrow) * SS + k);
        v2f w3 = *reinterpret_cast<const v2f*>(wu + (size_t)(nb + 48 + mrow) * SS + k);
        d0 = wmma_f32x4(areg[t], w0, d0);
        d1 = wmma_f32x4(areg[t], w1, d1);
        d2 = wmma_f32x4(areg[t], w2, d2);
        d3 = wmma_f32x4(areg[t], w3, d3);
      }
      v8f dsv[4] = {d0, d1, d2, d3};
#pragma unroll
      for (int q = 0; q < 4; ++q) {
        const int n = nb + q * 16 + mrow;
#pragma unroll
        for (int j = 0; j < 8; ++j) {
          const int pr = prs[j];
          if (pr >= 0) partial[(size_t)pr * DD + n] = dsv[q][j];
        }
      }
    }
  }
}

// ---------------------------------------------------------------------------
// K4 (fallback, no `partial` scratch): row-based VALU decode.
// ---------------------------------------------------------------------------
__global__ __launch_bounds__(256) void k4_decode(
    const float* __restrict__ W_rd, const float* __restrict__ W_up,
    const float* __restrict__ dec_bias, const int* __restrict__ top_idx,
    const float* __restrict__ subrec, float* __restrict__ out) {
  const int b = blockIdx.x;
  const int t = threadIdx.x;
  __shared__ float vsm[KTOP][SS];
  __shared__ int psm[KTOP];
  if (t < KTOP) psm[t] = top_idx[b * KTOP + t];
  for (int i = t; i < KTOP * SS; i += 256)
    vsm[i >> 6][i & (SS - 1)] = subrec[(size_t)b * KTOP * SS + i];
  __syncthreads();

#pragma unroll
  for (int dd = 0; dd < DD / 256; ++dd) {
    const int d = t + dd * 256;
    float acc = dec_bias[d];
#pragma unroll
    for (int j = 0; j < KTOP; ++j) {
      const int p = psm[j];
      acc += BETA * W_rd[(size_t)d * PP + p];
      const float4* wu4 = reinterpret_cast<const float4*>(
          W_up + ((size_t)p * DD + d) * SS);
      const float4* vv4 = reinterpret_cast<const float4*>(vsm[j]);
      float s = 0.f;
#pragma unroll
      for (int q = 0; q < SS / 4; ++q) {
        const float4 w = wu4[q];
        const float4 v = vv4[q];
        s += w.x * v.x + w.y * v.y + w.z * v.z + w.w * v.w;
      }
      acc += s;
    }
    out[(size_t)b * DD + d] = acc;
  }
}

// ---------------------------------------------------------------------------
// K4b (fast path): out[b,:] = dec_bias + sum_j (BETA*W_rd[:,p_j] + partial_j)
// Fixed summation order -> bitwise deterministic.
// ---------------------------------------------------------------------------
__global__ __launch_bounds__(256) void k4b_reduce(
    const float* __restrict__ W_rd, const float* __restrict__ dec_bias,
    const int* __restrict__ top_idx, const float* __restrict__ partial,
    float* __restrict__ out) {
  const int b = blockIdx.x;
  const int t = threadIdx.x;
  __shared__ int psm[KTOP];
  if (t < KTOP) psm[t] = top_idx[b * KTOP + t];
  __syncthreads();
#pragma unroll
  for (int dd = 0; dd < DD / 256; ++dd) {
    const int d = t + dd * 256;
    float acc = dec_bias[d];
#pragma unroll
    for (int j = 0; j < KTOP; ++j) {
      acc += BETA * W_rd[(size_t)d * PP + psm[j]];
      acc += partial[((size_t)b * KTOP + j) * DD + d];
    }
    out[(size_t)b * DD + d] = acc;
  }
}

// ---------------------------------------------------------------------------
extern "C" void kernel_launch(void* const* d_in, const int* in_sizes, int n_in,
                              void* d_out, int out_size, void* d_ws,
                              size_t ws_size, hipStream_t stream) {
  (void)in_sizes; (void)n_in; (void)out_size;
  const float* x        = (const float*)d_in[0];
  const float* W_re     = (const float*)d_in[1];
  const float* b_re     = (const float*)d_in[2];
  const float* W_rd     = (const float*)d_in[3];
  const float* W_down   = (const float*)d_in[4];
  const float* W_up     = (const float*)d_in[5];
  const float* W_ce     = (const float*)d_in[6];
  const float* b_ce     = (const float*)d_in[7];
  const float* W_cd     = (const float*)d_in[8];
  const float* dec_bias = (const float*)d_in[9];
  float* out = (float*)d_out;

  // Workspace carve-up: base ~16.2 MB; optional 256 MB pair-partial buffer.
  char* ws = (char*)d_ws;
  size_t used = 0;
  float* logits  = (float*)(ws + used); used += (size_t)BB * PP * sizeof(float);
  int*   rowlist = (int*)(ws + used);   used += (size_t)PP * BB * sizeof(int);
  float* subrec  = (float*)(ws + used); used += (size_t)BB * KTOP * SS * sizeof(float);
  int*   top_idx = (int*)(ws + used);   used += (size_t)BB * KTOP * sizeof(int);
  int*   counts  = (int*)(ws + used);   used += 256;  // 256 B = 64 ints (PP counters need 1 KB)
  const size_t partialBytes = (size_t)BB * KTOP * DD * sizeof(float);
  float* partial = nullptr;
  if (ws_size >= used + partialBytes) {   // fixed per harness run: deterministic
    partial = (float*)(ws + used);
    used += partialBytes;
  }

  k0_zero_counts<<<1, 256, 0, stream>>>(counts);
  k1_parent_logits<<<dim3(BB / 16, PP / 64), 32, 0, stream>>>(x, W_re, b_re,
                                                              logits);
  k2_topk_and_lists<<<BB / 256, 256, 0, stream>>>(logits, top_idx, counts,
                                                  rowlist);
  k3_child_encode<<<dim3(PP, BB / 16), 32, 0, stream>>>(
      x, W_down, W_ce, b_ce, W_cd, W_up, counts, rowlist, subrec, partial);
  if (partial != nullptr) {
    k4b_reduce<<<BB, 256, 0, stream>>>(W_rd, dec_bias, top_idx, partial, out);
  } else {
    k4_decode<<<BB, 256, 0, stream>>>(W_rd, W_up, dec_bias, top_idx, subrec,
                                      out);
  }
}